// SkeletonSimnn_80814104641806
// MI455X (gfx1250) — compile-verified
//
#include <hip/hip_runtime.h>
#include <math.h>

typedef __attribute__((ext_vector_type(2))) float v2f;
typedef __attribute__((ext_vector_type(4))) float v4f;
typedef __attribute__((ext_vector_type(8))) float v8f;

#define NROWS 8192
#define DIM   256
#define KNN   8
#define AST   260   // padded LDS stride for A tile (floats): 260 % 64 == 4 -> conflict-free
#define SST   17    // padded stride for 16x16 score scratch

// ---------------------------------------------------------------------------
// Kernel 1: L2-normalize rows. One wave (32 lanes) per 256-float row.
// ---------------------------------------------------------------------------
__global__ __launch_bounds__(256)
void normalize_rows(const float* __restrict__ src, float* __restrict__ dst) {
    const int wave = threadIdx.x >> 5;
    const int lane = threadIdx.x & 31;
    const int row  = blockIdx.x * 8 + wave;     // grid = NROWS/8
    const float* s = src + (size_t)row * DIM + lane * 8;
    float*       d = dst + (size_t)row * DIM + lane * 8;
    v4f a = *(const v4f*)(s);
    v4f b = *(const v4f*)(s + 4);
    float ss = a.x*a.x + a.y*a.y + a.z*a.z + a.w*a.w
             + b.x*b.x + b.y*b.y + b.z*b.z + b.w*b.w;
#pragma unroll
    for (int m = 16; m >= 1; m >>= 1) ss += __shfl_xor(ss, m, 32);
    const float inv = 1.0f / fmaxf(sqrtf(ss), 1e-12f);
    a *= inv; b *= inv;
    *(v4f*)(d)     = a;
    *(v4f*)(d + 4) = b;
}

// ---------------------------------------------------------------------------
// Kernel 2: fused similarity GEMM (fp32 WMMA) + streaming top-8 + gather.
// One workgroup (8 waves) per 16-row tile; each wave sweeps 64 col-tiles.
// ---------------------------------------------------------------------------
__global__ __launch_bounds__(256)
void mine_topk(const float* __restrict__ xh,
               const float* __restrict__ yh,
               const float* __restrict__ ypool_raw,
               const int*   __restrict__ sel,
               float*       __restrict__ out) {
    __shared__ __align__(16) float xhs[16 * AST];          // A tile, padded
    __shared__ __align__(16) float sc[8 * 16 * SST];       // per-wave 16x16 score scratch
    __shared__ float wval[8][16][KNN];                     // per-wave top-8 values
    __shared__ int   widx[8][16][KNN];                     // per-wave top-8 indices
    __shared__ int   ids[16];

    const int tid     = threadIdx.x;
    const int wave    = tid >> 5;
    const int lane    = tid & 31;
    const int rowBase = blockIdx.x * 16;

    // ---- stage 16x256 A tile into LDS (coalesced float4, padded rows) ----
    for (int i = tid; i < 16 * (DIM / 4); i += 256) {
        const int r = i >> 6;              // 0..15
        const int c = (i & 63) * 4;        // 0..252
        v4f v = *(const v4f*)(xh + (size_t)(rowBase + r) * DIM + c);
        *(v4f*)(&xhs[r * AST + c]) = v;
    }
    __syncthreads();

    // per-lane sorted top-8 (descending value); lane l<16 owns row l
    float tv[KNN];
    int   tj[KNN];
#pragma unroll
    for (int t = 0; t < KNN; ++t) { tv[t] = -INFINITY; tj[t] = 0x7fffffff; }

    float* myScore = &sc[wave * 16 * SST];
    const int m    = lane & 15;            // A row / B column within tile
    const int koff = (lane >> 4) * 2;      // K sub-offset per half-wave

    const float* abase = &xhs[m * AST + koff];

    for (int jt = wave; jt < NROWS / 16; jt += 8) {   // uniform per wave: 64 iters
        const int j0 = jt * 16;
        const float* bbase = yh + (size_t)(j0 + m) * DIM + koff;

        v8f acc = {};
#pragma unroll 8
        for (int k = 0; k < DIM; k += 4) {
            v2f af = *(const v2f*)(abase + k);   // ds_load_b64
            v2f bf = *(const v2f*)(bbase + k);   // global_load_b64 (L2-resident)
            acc = __builtin_amdgcn_wmma_f32_16x16x4_f32(
                    false, af, false, bf, (short)0, acc, false, false);
        }

        // spill 16x16 scores: acc[r] -> row (r + 8*(lane>>4)), col (lane&15)
        const int rr = (lane >> 4) * 8;
        const int cc = lane & 15;
#pragma unroll
        for (int r = 0; r < 8; ++r)
            myScore[(rr + r) * SST + cc] = acc[r];

        // lanes 0..15: row-owner streaming top-8 update (stable: tie->smaller j)
        if (lane < 16) {
#pragma unroll
            for (int c = 0; c < 16; ++c) {
                const float v = myScore[lane * SST + c];
                const int   j = j0 + c;
                if (v > tv[KNN - 1]) {
#pragma unroll
                    for (int t = KNN - 1; t >= 1; --t) {
                        const bool up = (v > tv[t - 1]);
                        const bool at = (v > tv[t]);
                        const float nv = up ? tv[t - 1] : v;
                        const int   nj = up ? tj[t - 1] : j;
                        tv[t] = at ? nv : tv[t];
                        tj[t] = at ? nj : tj[t];
                    }
                    if (v > tv[0]) { tv[0] = v; tj[0] = j; }
                }
            }
        }
    }

    // ---- publish per-wave lists ----
    if (lane < 16) {
#pragma unroll
        for (int t = 0; t < KNN; ++t) {
            wval[wave][lane][t] = tv[t];
            widx[wave][lane][t] = tj[t];
        }
    }
    __syncthreads();

    // ---- wave 0 merges 64 candidates/row with (value desc, index asc) order ----
    if (wave == 0 && lane < 16) {
        float mv[KNN];
        int   mj[KNN];
#pragma unroll
        for (int t = 0; t < KNN; ++t) { mv[t] = -INFINITY; mj[t] = 0x7fffffff; }
        for (int w = 0; w < 8; ++w) {
#pragma unroll
            for (int e = 0; e < KNN; ++e) {
                const float v = wval[w][lane][e];
                const int   j = widx[w][lane][e];
                const bool b7 = (v > mv[KNN - 1]) ||
                                (v == mv[KNN - 1] && j < mj[KNN - 1]);
                if (b7) {
#pragma unroll
                    for (int t = KNN - 1; t >= 1; --t) {
                        const bool up = (v > mv[t - 1]) || (v == mv[t - 1] && j < mj[t - 1]);
                        const bool at = (v > mv[t])     || (v == mv[t]     && j < mj[t]);
                        const float nv = up ? mv[t - 1] : v;
                        const int   nj = up ? mj[t - 1] : j;
                        mv[t] = at ? nv : mv[t];
                        mj[t] = at ? nj : mj[t];
                    }
                    const bool b0 = (v > mv[0]) || (v == mv[0] && j < mj[0]);
                    if (b0) { mv[0] = v; mj[0] = j; }
                }
            }
        }
        const int s = sel[rowBase + lane] & 7;   // sel in [0,8)
        ids[lane] = mj[s];
    }
    __syncthreads();

    // ---- gather mined rows from RAW y_pool: 16 threads per row, float4 ----
    {
        const int r    = tid >> 4;     // 0..15
        const int part = tid & 15;     // 0..15 -> 16 floats each
        const float* srow = ypool_raw + (size_t)ids[r] * DIM + part * 16;
        float*       drow = out + (size_t)(rowBase + r) * DIM + part * 16;
#pragma unroll
        for (int q = 0; q < 4; ++q)
            *(v4f*)(drow + q * 4) = *(const v4f*)(srow + q * 4);
    }
}

// ---------------------------------------------------------------------------
extern "C" void kernel_launch(void* const* d_in, const int* in_sizes, int n_in,
                              void* d_out, int out_size, void* d_ws, size_t ws_size,
                              hipStream_t stream) {
    (void)in_sizes; (void)n_in; (void)out_size; (void)ws_size;

    const float* y     = (const float*)d_in[0];
    const float* ypool = (const float*)d_in[1];
    const int*   sel   = (const int*)d_in[2];
    float*       out   = (float*)d_out;

    float* xh = (float*)d_ws;                       // 8192*256 f32 = 8 MB
    float* yh = xh + (size_t)NROWS * DIM;           // next 8 MB

    normalize_rows<<<NROWS / 8, 256, 0, stream>>>(y, xh);
    normalize_rows<<<NROWS / 8, 256, 0, stream>>>(ypool, yh);
    mine_topk<<<NROWS / 16, 256, 0, stream>>>(xh, yh, ypool, sel, out);
}